// EnvMap_BRDF_renderLoss_15144054686501
// MI455X (gfx1250) — compile-verified
//
#include <hip/hip_runtime.h>
#include <math.h>

// ---------------- problem constants (from reference) ----------------
#define IM   256
#define NP   (IM * IM)          // 65536 pixels
#define EXW  256
#define EXH  128
#define ENVN (EXH * EXW * 3)    // elems per env image (one batch)
#define S    128
#define NBLK 256                // 256 blocks x 256 threads = 65536 pixels
#define F0c  0.05f
#define PI_F 3.14159265358979f
#define FOUR_PI 12.566370614359172f

typedef float v2f __attribute__((ext_vector_type(2)));
typedef float v8f __attribute__((ext_vector_type(8)));

__device__ __forceinline__ float clamp01(float a) {
    return fminf(fmaxf(a, 0.0f), 1.0f);
}

// ------------------------------------------------------------------
// Main kernel: one thread per pixel. Loops over 128 light samples,
// gathers 4 env images (x[0],x[1],y[0],y[1]) at shared indices,
// accumulates diffuse/spec sums, writes pred[0]/gt[0], and emits
// per-block partial sums of the squared-error loss and of seg.
// ------------------------------------------------------------------
__global__ __launch_bounds__(256) void env_brdf_kernel(
    const float* __restrict__ x,      // (2,128,256,3)
    const float* __restrict__ y,      // (2,128,256,3)
    const float* __restrict__ diffuse,// (3,256,256)
    const float* __restrict__ normal, // (3,256,256)
    const float* __restrict__ rough,  // (1,256,256)
    const float* __restrict__ seg,    // (256,256)
    const float* __restrict__ v,      // (3,256,256)
    const float* __restrict__ ls,     // (1,128,3)
    const float* __restrict__ ew,     // (1,128,1)
    const int*   __restrict__ idy,    // (128,256,256)
    const int*   __restrict__ idx,    // (128,256,256)
    float* __restrict__ out,          // [0]=loss, [1..]=pred0 (3,H,W), then gt0
    float* __restrict__ partial_loss, // 256 floats (workspace)
    float* __restrict__ partial_seg)  // 256 floats (workspace)
{
    __shared__ float s_ls[S * 3];
    __shared__ float s_ew[S];
    __shared__ float rl[256];
    __shared__ float rs[256];

    const int t = threadIdx.x;
    const int p = blockIdx.x * 256 + t;   // pixel linear index (h*256+w)

    // stage small constants into LDS (reused 128x per thread)
    for (int i = t; i < S * 3; i += 256) s_ls[i] = ls[i];
    if (t < S) s_ew[t] = ew[t];
    __syncthreads();

    // ---- per-pixel setup ----
    const float nx = normal[p], ny = normal[NP + p], nz = normal[2 * NP + p];
    const float vx = v[p],      vy = v[NP + p],      vz = v[2 * NP + p];

    // camy = normalize(up - (up.n) n), up = (0,1,0)
    float cyx = -ny * nx, cyy = 1.0f - ny * ny, cyz = -ny * nz;
    float cl  = sqrtf(cyx * cyx + cyy * cyy + cyz * cyz);
    cl = fmaxf(cl, 1e-12f);
    const float icl = 1.0f / cl;
    cyx *= icl; cyy *= icl; cyz *= icl;

    // camx = -(camy x normal) / L1norm
    float cxx = cyy * nz - cyz * ny;
    float cxy = cyz * nx - cyx * nz;
    float cxz = cyx * ny - cyy * nx;
    float l1  = fabsf(cxx) + fabsf(cxy) + fabsf(cxz);
    l1 = fmaxf(l1, 1e-12f);
    const float inl1 = -1.0f / l1;
    cxx *= inl1; cxy *= inl1; cxz *= inl1;

    const float roughB = (rough[p] + 1.0f) * 0.5f;
    const float kk     = (roughB + 1.0f) * (roughB + 1.0f) * 0.125f;
    float a2 = roughB * roughB; a2 = a2 * a2;
    const float ndv  = clamp01(nx * vx + ny * vy + nz * vz);
    const float nom1 = ndv * (1.0f - kk) + kk;
    const float a2m1 = a2 - 1.0f;
    const float omk  = 1.0f - kk;

    const float* envp0 = x;            // x batch 0
    const float* envp1 = x + ENVN;     // x batch 1
    const float* envp2 = y;            // y batch 0
    const float* envp3 = y + ENVN;     // y batch 1

    float aD[4][3] = {{0.f,0.f,0.f},{0.f,0.f,0.f},{0.f,0.f,0.f},{0.f,0.f,0.f}};
    float aS[4][3] = {{0.f,0.f,0.f},{0.f,0.f,0.f},{0.f,0.f,0.f},{0.f,0.f,0.f}};

    for (int s = 0; s < S; ++s) {
        // prefetch the index stream a few samples ahead (global_prefetch_b8)
        if (s + 4 < S) {
            __builtin_prefetch(&idy[(s + 4) * NP + p], 0, 0);
            __builtin_prefetch(&idx[(s + 4) * NP + p], 0, 0);
        }

        const float lsx = s_ls[3 * s + 0];
        const float lsy = s_ls[3 * s + 1];
        const float lsz = s_ls[3 * s + 2];

        // l = ls.x*camx + ls.y*camy + ls.z*normal
        const float lx = lsx * cxx + lsy * cyx + lsz * nx;
        const float ly = lsx * cxy + lsy * cyy + lsz * ny;
        const float lz = lsx * cxz + lsy * cyz + lsz * nz;

        // h = normalize((v+l)/2), with clip on |h|^2
        float hx = (vx + lx) * 0.5f, hy = (vy + ly) * 0.5f, hz = (vz + lz) * 0.5f;
        float hh = hx * hx + hy * hy + hz * hz;
        hh = fmaxf(hh, 1e-6f);
        const float rin = rsqrtf(hh);
        hx *= rin; hy *= rin; hz *= rin;

        const float vdh   = vx * hx + vy * hy + vz * hz;
        const float frac0 = F0c + (1.0f - F0c) *
                            exp2f((-5.55472f * vdh - 6.98316f) * vdh);

        const float ndh = clamp01(nx * hx + ny * hy + nz * hz);
        const float ndl = clamp01(nx * lx + ny * ly + nz * lz);

        const float nom0 = ndh * ndh * a2m1 + 1.0f;
        const float nom2 = ndl * omk + kk;
        float nom = FOUR_PI * nom0 * nom0 * nom1 * nom2;
        nom = fminf(fmaxf(nom, 1e-6f), FOUR_PI);
        const float spec = a2 * frac0 / nom;

        const float w  = s_ew[s];
        const float cD = ndl * w;
        const float cS = spec * cD;

        const int iy   = idy[s * NP + p];
        const int ix   = idx[s * NP + p];
        const int base = (iy * EXW + ix) * 3;

        {
            const float e0 = envp0[base], e1 = envp0[base + 1], e2 = envp0[base + 2];
            aD[0][0] += cD * e0; aD[0][1] += cD * e1; aD[0][2] += cD * e2;
            aS[0][0] += cS * e0; aS[0][1] += cS * e1; aS[0][2] += cS * e2;
        }
        {
            const float e0 = envp1[base], e1 = envp1[base + 1], e2 = envp1[base + 2];
            aD[1][0] += cD * e0; aD[1][1] += cD * e1; aD[1][2] += cD * e2;
            aS[1][0] += cS * e0; aS[1][1] += cS * e1; aS[1][2] += cS * e2;
        }
        {
            const float e0 = envp2[base], e1 = envp2[base + 1], e2 = envp2[base + 2];
            aD[2][0] += cD * e0; aD[2][1] += cD * e1; aD[2][2] += cD * e2;
            aS[2][0] += cS * e0; aS[2][1] += cS * e1; aS[2][2] += cS * e2;
        }
        {
            const float e0 = envp3[base], e1 = envp3[base + 1], e2 = envp3[base + 2];
            aD[3][0] += cD * e0; aD[3][1] += cD * e1; aD[3][2] += cD * e2;
            aS[3][0] += cS * e0; aS[3][1] += cS * e1; aS[3][2] += cS * e2;
        }
    }

    // ---- finalize: pred = diffB*sumD + 10*sumS ----
    const float inv2pi = 0.5f / PI_F;
    float diffB[3];
    diffB[0] = (diffuse[p] + 1.0f) * inv2pi;
    diffB[1] = (diffuse[NP + p] + 1.0f) * inv2pi;
    diffB[2] = (diffuse[2 * NP + p] + 1.0f) * inv2pi;

    float pr[2][3], gt[2][3];
#pragma unroll
    for (int c = 0; c < 3; ++c) {
        pr[0][c] = diffB[c] * aD[0][c] + 10.0f * aS[0][c];
        pr[1][c] = diffB[c] * aD[1][c] + 10.0f * aS[1][c];
        gt[0][c] = diffB[c] * aD[2][c] + 10.0f * aS[2][c];
        gt[1][c] = diffB[c] * aD[3][c] + 10.0f * aS[3][c];
    }

    // outputs: [0]=loss, then pred[0] (3,H,W), then gt[0] (3,H,W)
#pragma unroll
    for (int c = 0; c < 3; ++c) {
        out[1 + c * NP + p]          = pr[0][c];
        out[1 + 3 * NP + c * NP + p] = gt[0][c];
    }

    // local squared-error contribution (both batches, 3 channels)
    float lsum = 0.0f;
#pragma unroll
    for (int c = 0; c < 3; ++c) {
        float d0 = pr[0][c] - gt[0][c];
        float d1 = pr[1][c] - gt[1][c];
        lsum += d0 * d0 + d1 * d1;
    }

    // deterministic in-block tree reduction of loss and seg
    rl[t] = lsum;
    rs[t] = seg[p];
    __syncthreads();
#pragma unroll
    for (int off = 128; off > 0; off >>= 1) {
        if (t < off) { rl[t] += rl[t + off]; rs[t] += rs[t + off]; }
        __syncthreads();
    }
    if (t == 0) {
        partial_loss[blockIdx.x] = rl[0];
        partial_seg[blockIdx.x]  = rs[0];
    }
}

// ------------------------------------------------------------------
// Final reduction of 256 loss partials + 256 seg partials by a single
// wave using V_WMMA_F32_16X16X4_F32 with B = all-ones:
//   D(+=C)[i][j] = sum_k A[i][k]  -> row sums; 4 accumulating WMMAs
// consume all 256 values (each multiplied by 1 exactly once), so the
// grand total is independent of the A-slot mapping. Column 0 of D is
// spread over lanes 0 (rows 0-7) and 16 (rows 8-15).
// ------------------------------------------------------------------
__global__ __launch_bounds__(32) void loss_reduce_kernel(
    const float* __restrict__ partial_loss,
    const float* __restrict__ partial_seg,
    float* __restrict__ out)
{
    const int lane = threadIdx.x;

    v8f cL = {0.f,0.f,0.f,0.f,0.f,0.f,0.f,0.f};
    v8f cS = {0.f,0.f,0.f,0.f,0.f,0.f,0.f,0.f};
    v2f ones; ones.x = 1.0f; ones.y = 1.0f;

#pragma unroll
    for (int j = 0; j < 4; ++j) {
        v2f aL, aS;
        aL.x = partial_loss[j * 64 + lane * 2];
        aL.y = partial_loss[j * 64 + lane * 2 + 1];
        aS.x = partial_seg [j * 64 + lane * 2];
        aS.y = partial_seg [j * 64 + lane * 2 + 1];
        // 8-arg pattern: (neg_a, A, neg_b, B, c_mod, C, reuse_a, reuse_b)
        cL = __builtin_amdgcn_wmma_f32_16x16x4_f32(
                false, aL, false, ones, (short)0, cL, false, false);
        cS = __builtin_amdgcn_wmma_f32_16x16x4_f32(
                false, aS, false, ones, (short)0, cS, false, false);
    }

    // per-lane sum over the 8 D registers:
    //   lanes 0-15  -> sum of row-sums for rows 0..7  (any column, identical)
    //   lanes 16-31 -> sum of row-sums for rows 8..15
    float sL = 0.0f, sS = 0.0f;
#pragma unroll
    for (int r = 0; r < 8; ++r) { sL += cL[r]; sS += cS[r]; }

    const float totL = __shfl(sL, 0, 32) + __shfl(sL, 16, 32);
    const float totS = __shfl(sS, 0, 32) + __shfl(sS, 16, 32);

    if (lane == 0) {
        // pixel_num = seg.sum() * B(=2) * 3
        out[0] = totL / (totS * 6.0f);
    }
}

// ------------------------------------------------------------------
extern "C" void kernel_launch(void* const* d_in, const int* in_sizes, int n_in,
                              void* d_out, int out_size, void* d_ws, size_t ws_size,
                              hipStream_t stream) {
    const float* x       = (const float*)d_in[0];
    const float* y       = (const float*)d_in[1];
    const float* diffuse = (const float*)d_in[2];
    const float* normal  = (const float*)d_in[3];
    const float* rough   = (const float*)d_in[4];
    const float* seg     = (const float*)d_in[5];
    const float* v       = (const float*)d_in[6];
    const float* ls      = (const float*)d_in[7];
    const float* ew      = (const float*)d_in[8];
    const int*   idy     = (const int*)d_in[9];
    const int*   idx     = (const int*)d_in[10];

    float* out = (float*)d_out;
    float* partial_loss = (float*)d_ws;         // 256 floats
    float* partial_seg  = partial_loss + 256;   // 256 floats

    env_brdf_kernel<<<NBLK, 256, 0, stream>>>(
        x, y, diffuse, normal, rough, seg, v, ls, ew, idy, idx,
        out, partial_loss, partial_seg);

    loss_reduce_kernel<<<1, 32, 0, stream>>>(partial_loss, partial_seg, out);
}